// OptimizedASOSENetwork_50895362457843
// MI455X (gfx1250) — compile-verified
//
#include <hip/hip_runtime.h>
#include <hip/hip_bf16.h>

typedef __attribute__((ext_vector_type(2))) float v2f;
typedef __attribute__((ext_vector_type(8))) float v8f;
typedef __attribute__((ext_vector_type(4))) unsigned int u32x4;
typedef __attribute__((ext_vector_type(4))) int i32x4;
typedef __attribute__((ext_vector_type(8))) int i32x8;

#define N_IMG 32

// ---------------------------------------------------------------------------
// TDM 2D tile load: global (tile1 rows x tile0 contiguous f32, row stride
// stride0 elements) -> LDS at byte offset lds_off.  Descriptor per
// CDNA5 ISA: group0 {count=1, lds_addr, global_addr,
// type=2}, group1 {data_size=4B, tensor_dims, tile_dims, dim0 stride},
// groups 2/3 zero (2D: tile_dim2/3 = 0).  Tracked by TENSORcnt.
// ---------------------------------------------------------------------------
__device__ __forceinline__ void tdm_load_2d(const float* gbase, unsigned lds_off,
                                            unsigned tile0, unsigned tile1,
                                            unsigned long long stride0_elems) {
  unsigned long long ga = (unsigned long long)(uintptr_t)gbase;
  u32x4 g0;
  g0[0] = 1u;                                               // count=1, user mode
  g0[1] = lds_off;                                          // lds_addr (bytes)
  g0[2] = (unsigned)(ga & 0xFFFFFFFFu);                     // global_addr[31:0]
  g0[3] = (unsigned)((ga >> 32) & 0x01FFFFFFu) | (2u << 30);// addr[56:32] | type=2
  i32x8 g1;
  g1[0] = 0x00020000;                                       // data_size=2 (4B)
  g1[1] = (int)((tile0 & 0xFFFFu) << 16);                   // tensor_dim0[15:0]
  g1[2] = (int)((tile0 >> 16) | ((tile1 & 0xFFFFu) << 16)); // tdim0 hi | tdim1 lo
  g1[3] = (int)((tile1 >> 16) | (tile0 << 16));             // tdim1 hi | tile_dim0
  g1[4] = (int)(tile1 & 0xFFFFu);                           // tile_dim1 (tile_dim2=0)
  g1[5] = (int)(stride0_elems & 0xFFFFFFFFull);             // dim0 stride lo
  g1[6] = (int)((stride0_elems >> 32) & 0xFFFFull);         // dim0 stride hi
  g1[7] = 0;
  i32x4 z4 = {0, 0, 0, 0};
#if __clang_major__ >= 23
  i32x8 z8 = {0, 0, 0, 0, 0, 0, 0, 0};
  __builtin_amdgcn_tensor_load_to_lds(g0, g1, z4, z4, z8, 0);
#else
  __builtin_amdgcn_tensor_load_to_lds(g0, g1, z4, z4, 0);
#endif
}

// ---------------------------------------------------------------------------
// Gate computation: softmax over 8 ops, top-3 above 0.01, renormalize.
// ---------------------------------------------------------------------------
__global__ void gates_kernel(const float* __restrict__ ap, float* __restrict__ g) {
  if (threadIdx.x != 0 || blockIdx.x != 0) return;
  for (int l = 0; l < 4; ++l) {
    const float* r = ap + l * 8;
    float m = r[0];
    for (int i = 1; i < 8; ++i) m = fmaxf(m, r[i]);
    float w[8], s = 0.f;
    for (int i = 0; i < 8; ++i) { w[i] = __expf(r[i] - m); s += w[i]; }
    for (int i = 0; i < 8; ++i) w[i] /= s;
    int idx[3]; float v[3];
    bool used[8] = {false,false,false,false,false,false,false,false};
    for (int k = 0; k < 3; ++k) {
      int best = 0; float bv = -1.f;
      for (int i = 0; i < 8; ++i)
        if (!used[i] && w[i] > bv) { bv = w[i]; best = i; }
      used[best] = true; idx[k] = best; v[k] = bv;
    }
    float sv = 0.f;
    for (int k = 0; k < 3; ++k) { v[k] = (v[k] > 0.01f) ? v[k] : 0.f; sv += v[k]; }
    float inv = 1.f / (sv + 1e-8f);
    float go[8] = {0,0,0,0,0,0,0,0};
    for (int k = 0; k < 3; ++k) go[idx[k]] = v[k] * inv;
    for (int i = 0; i < 8; ++i) g[l * 8 + i] = go[i];
  }
}

// ---------------------------------------------------------------------------
// Stem: 3x3 conv 3->32, pad 1, + ReLU.
// ---------------------------------------------------------------------------
__global__ void stem_kernel(const float* __restrict__ x, const float* __restrict__ wt,
                            float* __restrict__ out) {
  int idx = blockIdx.x * blockDim.x + threadIdx.x;
  int w = idx & 127; int t = idx >> 7;
  int h = t & 127;   t >>= 7;
  int co = t & 31;   int n = t >> 5;
  float acc = 0.f;
  for (int ci = 0; ci < 3; ++ci) {
    const float* xp = x + ((n * 3 + ci) * 128) * 128;
    const float* wp = wt + (co * 3 + ci) * 9;
    for (int kh = 0; kh < 3; ++kh) {
      int ih = h + kh - 1; if ((unsigned)ih >= 128u) continue;
      for (int kw = 0; kw < 3; ++kw) {
        int iw = w + kw - 1; if ((unsigned)iw >= 128u) continue;
        acc += xp[ih * 128 + iw] * wp[kh * 3 + kw];
      }
    }
  }
  out[idx] = fmaxf(acc, 0.f);
}

// ---------------------------------------------------------------------------
// Depthwise conv with fused input-ReLU; gated early-exit.
// ---------------------------------------------------------------------------
__global__ void dw_relu_kernel(const float* __restrict__ in, const float* __restrict__ wt,
                               float* __restrict__ out, int C, int H, int W,
                               int K, int pad, int dil,
                               const float* __restrict__ gates, int gidx) {
  if (gates && gates[gidx] == 0.f) return;
  int idx = blockIdx.x * blockDim.x + threadIdx.x;
  int w = idx % W; int t = idx / W;
  int h = t % H;   t /= H;
  int c = t % C;   int n = t / C;
  const float* ip = in + (n * C + c) * H * W;
  const float* wp = wt + c * K * K;
  float acc = 0.f;
  for (int kh = 0; kh < K; ++kh) {
    int ih = h + kh * dil - pad; if ((unsigned)ih >= (unsigned)H) continue;
    for (int kw = 0; kw < K; ++kw) {
      int iw = w + kw * dil - pad; if ((unsigned)iw >= (unsigned)W) continue;
      acc += fmaxf(ip[ih * W + iw], 0.f) * wp[kh * K + kw];
    }
  }
  out[idx] = acc;
}

// ---------------------------------------------------------------------------
// Fused  y = g1*maxpool3 + g2*avgpool3(count_exclude_pad) + g3*skip.
// ---------------------------------------------------------------------------
__global__ void pool_skip_kernel(const float* __restrict__ x, float* __restrict__ y,
                                 int C, int H, int W, const float* __restrict__ g) {
  float g1 = g[1], g2 = g[2], g3 = g[3];
  int idx = blockIdx.x * blockDim.x + threadIdx.x;
  int w = idx % W; int t = idx / W;
  int h = t % H;   t /= H;
  const float* p = x + t * H * W;
  int h0 = h > 0 ? h - 1 : 0, h1 = h < H - 1 ? h + 1 : H - 1;
  int w0 = w > 0 ? w - 1 : 0, w1 = w < W - 1 ? w + 1 : W - 1;
  float res = 0.f;
  if (g1 != 0.f) {
    float m = -3.4e38f;
    for (int hh = h0; hh <= h1; ++hh)
      for (int ww = w0; ww <= w1; ++ww) m = fmaxf(m, p[hh * W + ww]);
    res += g1 * m;
  }
  if (g2 != 0.f) {
    float s = 0.f;
    for (int hh = h0; hh <= h1; ++hh)
      for (int ww = w0; ww <= w1; ++ww) s += p[hh * W + ww];
    res += g2 * s / (float)((h1 - h0 + 1) * (w1 - w0 + 1));
  }
  if (g3 != 0.f) res += g3 * p[h * W + w];
  y[idx] = res;
}

// ---------------------------------------------------------------------------
// Stride-1 pointwise conv GEMM with TDM->LDS double-buffered B staging.
// out[Co,P] = W[Co,Ci] x act[Ci,P], P = 32*HW.
// Block = 8 waves = (Cout/16 co-tiles) x (pPerBlock p-tiles of 64 cols).
// Wave 0 issues TENSOR_LOAD_TO_LDS for the next 16-channel slab while all
// waves run V_WMMA_F32_16X16X4_F32 on the current slab from LDS.
// mode 0: write    mode 2: out += gate*result
// ---------------------------------------------------------------------------
__global__ void pw_wmma_lds_kernel(const float* __restrict__ in,
                                   const float* __restrict__ wt,
                                   float* __restrict__ out,
                                   int Cin, int Cout, int HW, int mode,
                                   const float* __restrict__ gates, int gidx) {
  float gval = 1.f;
  if (gates) {
    float gv = gates[gidx];
    if (gv == 0.f) return;                 // grid-uniform early exit
    if (mode == 2) gval = gv;
  }
  __shared__ float lbuf[2][4 * 16 * 64];   // [buf][pp][ch][col] : 32 KB

  int coTiles   = Cout >> 4;               // 2 / 4 / 8
  int pPerBlock = 8 / coTiles;             // 4 / 2 / 1
  int tid  = threadIdx.x;
  int wave = tid >> 5;
  int lane = tid & 31;
  int half = lane >> 4, lq = lane & 15;
  int co_t = wave % coTiles;
  int pp   = wave / coTiles;
  int co0  = co_t << 4;
  int nSlabs = Cin >> 4;

  // per-wave output column bookkeeping (64 cols, contiguous, single image)
  int colBase = (blockIdx.x * pPerBlock + pp) * 64;
  int n_img = colBase / HW;
  int s0    = colBase - n_img * HW;
  int offOut = n_img * Cout * HW + s0;     // + m*HW + col later

  // wave 0: issue slab 0 for every p-tile in this block
  if (wave == 0) {
    for (int q = 0; q < pPerBlock; ++q) {
      int cb = (blockIdx.x * pPerBlock + q) * 64;
      int nq = cb / HW; int sq = cb - nq * HW;
      const float* gb = in + (size_t)nq * Cin * HW + sq;     // slab 0, ch 0
      unsigned loff = (unsigned)(uintptr_t)&lbuf[0][q * 1024];
      tdm_load_2d(gb, loff, 64u, 16u, (unsigned long long)HW);
    }
    __builtin_amdgcn_s_wait_tensorcnt(0);
  }
  __syncthreads();

  v8f zero = {0.f,0.f,0.f,0.f,0.f,0.f,0.f,0.f};
  v8f acc[4];
#pragma unroll
  for (int j = 0; j < 4; ++j) acc[j] = zero;

  const float* wrow = wt + (co0 + lq) * Cin + (half << 1);
  int buf = 0;
  for (int slab = 0; slab < nSlabs; ++slab) {
    // prefetch next slab into the other buffer while we compute
    if (wave == 0 && slab + 1 < nSlabs) {
      for (int q = 0; q < pPerBlock; ++q) {
        int cb = (blockIdx.x * pPerBlock + q) * 64;
        int nq = cb / HW; int sq = cb - nq * HW;
        const float* gb = in + (size_t)nq * Cin * HW + (size_t)(slab + 1) * 16 * HW + sq;
        unsigned loff = (unsigned)(uintptr_t)&lbuf[buf ^ 1][q * 1024];
        tdm_load_2d(gb, loff, 64u, 16u, (unsigned long long)HW);
      }
    }
    const float* L = &lbuf[buf][pp * 1024];
#pragma unroll
    for (int ci = 0; ci < 16; ci += 4) {
      v2f a;
      a.x = wrow[slab * 16 + ci];
      a.y = wrow[slab * 16 + ci + 1];
      int ch = ci + (half << 1);
#pragma unroll
      for (int j = 0; j < 4; ++j) {
        const float* bp = L + ch * 64 + (j << 4) + lq;
        v2f b; b.x = bp[0]; b.y = bp[64];
        acc[j] = __builtin_amdgcn_wmma_f32_16x16x4_f32(
            false, a, false, b, (short)0, acc[j], false, false);
      }
    }
    __syncthreads();                       // everyone done reading lbuf[buf]
    if (wave == 0 && slab + 1 < nSlabs)
      __builtin_amdgcn_s_wait_tensorcnt(0);// next slab landed in lbuf[buf^1]
    __syncthreads();
    buf ^= 1;
  }

#pragma unroll
  for (int j = 0; j < 4; ++j) {
#pragma unroll
    for (int r = 0; r < 8; ++r) {
      int m = co0 + (half << 3) + r;
      int o = offOut + m * HW + (j << 4) + lq;
      float v = acc[j][r] * gval;
      if (mode == 2) out[o] += v;          // unique writer per element
      else           out[o] = v;
    }
  }
}

// ---------------------------------------------------------------------------
// Direct-global WMMA pointwise conv (handles stride != 1): used for the two
// strided 1x1 in_convs only.  mode 1: write+ReLU.
// ---------------------------------------------------------------------------
__global__ void pw_wmma_direct_kernel(const float* __restrict__ in,
                                      const float* __restrict__ wt,
                                      float* __restrict__ out,
                                      int Cin, int Cout, int Hin, int Win,
                                      int Hout, int Wout, int stride) {
  int HWin = Hin * Win, HWout = Hout * Wout;
  int wave = (int)((blockIdx.x * blockDim.x + threadIdx.x) >> 5);
  int lane = threadIdx.x & 31;
  int coTiles = Cout >> 4;
  int pTiles  = (N_IMG * HWout) >> 6;
  if (wave >= coTiles * pTiles) return;    // wave-uniform: EXEC stays all-ones
  int co_t = wave % coTiles;
  int p_t  = wave / coTiles;
  int half = lane >> 4, lq = lane & 15;
  int co0 = co_t << 4;

  int offIn[4], offOut[4];
#pragma unroll
  for (int j = 0; j < 4; ++j) {
    int col = (p_t << 6) + (j << 4) + lq;
    int n = col / HWout; int s = col - n * HWout;
    int hh = s / Wout; int ww = s - hh * Wout;
    offIn[j]  = n * Cin * HWin + hh * stride * Win + ww * stride;
    offOut[j] = n * Cout * HWout + s;
  }

  v8f zero = {0.f,0.f,0.f,0.f,0.f,0.f,0.f,0.f};
  v8f acc[4];
#pragma unroll
  for (int j = 0; j < 4; ++j) acc[j] = zero;

  const float* wrow = wt + (co0 + lq) * Cin + (half << 1);
  for (int ci = 0; ci < Cin; ci += 4) {
    v2f a; a.x = wrow[ci]; a.y = wrow[ci + 1];
    int cb = ci + (half << 1);
#pragma unroll
    for (int j = 0; j < 4; ++j) {
      const float* bp = in + offIn[j] + cb * HWin;
      v2f b; b.x = bp[0]; b.y = bp[HWin];
      acc[j] = __builtin_amdgcn_wmma_f32_16x16x4_f32(
          false, a, false, b, (short)0, acc[j], false, false);
    }
  }
#pragma unroll
  for (int j = 0; j < 4; ++j)
#pragma unroll
    for (int r = 0; r < 8; ++r) {
      int m = co0 + (half << 3) + r;
      out[offOut[j] + m * HWout] = fmaxf(acc[j][r], 0.f);   // in_conv: ReLU
    }
}

// ---------------------------------------------------------------------------
// Global average pool (128ch, 32x32) + linear 128->10.
// ---------------------------------------------------------------------------
__global__ void head_kernel(const float* __restrict__ x, const float* __restrict__ cw,
                            const float* __restrict__ cb, float* __restrict__ out) {
  __shared__ float sm[128];
  int n = blockIdx.x, c = threadIdx.x;
  const float* p = x + (n * 128 + c) * 1024;
  float s = 0.f;
  for (int i = 0; i < 1024; ++i) s += p[i];
  sm[c] = s * (1.f / 1024.f);
  __syncthreads();
  if (c < 10) {
    float acc = cb[c];
    for (int k = 0; k < 128; ++k) acc += sm[k] * cw[c * 128 + k];
    out[n * 10 + c] = acc;
  }
}

// ---------------------------------------------------------------------------
// Host driver.  Input order = JAX pytree leaves (dict keys sorted, None
// dropped): [0]=arch_params [1]=cls_b [2]=cls_w, per layer
// (dil3 d/pw, dil5 d/pw, in_conv?, sep3 d1/p1/d2/p2, sep5 d1/p1/d2/p2),
// bases 3/15/28/41, in_conv at 19 and 32, [53]=stem, [54]=x.
// Requires ws_size >= ~295 MB.
// ---------------------------------------------------------------------------
extern "C" void kernel_launch(void* const* d_in, const int* in_sizes, int n_in,
                              void* d_out, int out_size, void* d_ws, size_t ws_size,
                              hipStream_t stream) {
  const float* arch   = (const float*)d_in[0];
  const float* cls_b  = (const float*)d_in[1];
  const float* cls_w  = (const float*)d_in[2];
  const float* stem_w = (const float*)d_in[53];
  const float* x_in   = (const float*)d_in[54];

  float* ws = (float*)d_ws;
  float* G  = ws;                       // 64 floats (32 used)
  float* BX = ws + 64;                  // 16M floats
  float* BY = BX + (size_t)(1u << 24);  // 16M
  float* BC = BY + (size_t)(1u << 24);  // 9M (in_conv out)
  float* T1 = BC + (size_t)(9u << 20);  // 16M
  float* T2 = T1 + (size_t)(1u << 24);  // 16M

  gates_kernel<<<1, 32, 0, stream>>>(arch, G);
  stem_kernel<<<(N_IMG * 32 * 128 * 128) / 256, 256, 0, stream>>>(x_in, stem_w, BX);

  const int CinL[4]  = {32, 32, 64, 128};
  const int CoutL[4] = {32, 64, 128, 128};
  const int HinL[4]  = {128, 128, 64, 32};
  const int HopL[4]  = {128, 64, 32, 32};
  const int strdL[4] = {1, 2, 2, 1};
  const int baseL[4] = {3, 15, 28, 41};
  const int icIdx[4] = {-1, 19, 32, -1};

  float* xb = BX;
  float* yb = BY;
  for (int l = 0; l < 4; ++l) {
    int C = CoutL[l], H = HopL[l], HW = H * H;
    int b = baseL[l];
    int o = (icIdx[l] >= 0) ? 1 : 0;
    const float* dil3d  = (const float*)d_in[b + 0];
    const float* dil3p  = (const float*)d_in[b + 1];
    const float* dil5d  = (const float*)d_in[b + 2];
    const float* dil5p  = (const float*)d_in[b + 3];
    const float* sep3d1 = (const float*)d_in[b + 4 + o];
    const float* sep3p1 = (const float*)d_in[b + 5 + o];
    const float* sep3d2 = (const float*)d_in[b + 6 + o];
    const float* sep3p2 = (const float*)d_in[b + 7 + o];
    const float* sep5d1 = (const float*)d_in[b + 8 + o];
    const float* sep5p1 = (const float*)d_in[b + 9 + o];
    const float* sep5d2 = (const float*)d_in[b + 10 + o];
    const float* sep5p2 = (const float*)d_in[b + 11 + o];

    const float* xc = xb;
    if (icIdx[l] >= 0) {  // strided 1x1 conv + ReLU (direct WMMA path)
      const float* wi = (const float*)d_in[icIdx[l]];
      int waves = (C >> 4) * ((N_IMG * HW) >> 6);
      pw_wmma_direct_kernel<<<(waves + 7) / 8, 256, 0, stream>>>(
          xb, wi, BC, CinL[l], C, HinL[l], HinL[l], H, H, strdL[l]);
      xc = BC;
    }

    int nelem = N_IMG * C * HW;
    pool_skip_kernel<<<nelem / 256, 256, 0, stream>>>(xc, yb, C, H, H, G + l * 8);

    // TDM-staged WMMA GEMM: grid = p-tiles / pPerBlock blocks of 8 waves
    auto PW = [&](const float* in, const float* wt, float* outp, int md, int gi) {
      int pPerBlock = 8 / (C >> 4);
      int blocks = ((N_IMG * HW) >> 6) / pPerBlock;
      pw_wmma_lds_kernel<<<blocks, 256, 0, stream>>>(in, wt, outp, C, C, HW, md, G, gi);
    };
    auto DW = [&](const float* in, const float* wt, float* outp, int K, int pad,
                  int dil, int gi) {
      dw_relu_kernel<<<nelem / 256, 256, 0, stream>>>(in, wt, outp, C, H, H, K,
                                                      pad, dil, G, gi);
    };
    int gb = l * 8;
    // sep_conv_3x3 (gate 4)
    DW(xc, sep3d1, T1, 3, 1, 1, gb + 4);
    PW(T1, sep3p1, T2, 0, gb + 4);
    DW(T2, sep3d2, T1, 3, 1, 1, gb + 4);
    PW(T1, sep3p2, yb, 2, gb + 4);
    // sep_conv_5x5 (gate 5)
    DW(xc, sep5d1, T1, 5, 2, 1, gb + 5);
    PW(T1, sep5p1, T2, 0, gb + 5);
    DW(T2, sep5d2, T1, 5, 2, 1, gb + 5);
    PW(T1, sep5p2, yb, 2, gb + 5);
    // dil_conv_3x3 (gate 6)
    DW(xc, dil3d, T1, 3, 2, 2, gb + 6);
    PW(T1, dil3p, yb, 2, gb + 6);
    // dil_conv_5x5 (gate 7)
    DW(xc, dil5d, T1, 5, 4, 2, gb + 7);
    PW(T1, dil5p, yb, 2, gb + 7);

    float* t = xb; xb = yb; yb = t;
  }

  head_kernel<<<N_IMG, 128, 0, stream>>>(xb, cls_w, cls_b, (float*)d_out);
}